// SpatialVolumeNet_79130477462114
// MI455X (gfx1250) — compile-verified
//
#include <hip/hip_runtime.h>
#include <math.h>

#define V   64
#define HW  256
#define SQ2H 0.70710678f          /* sin/cos of radians(45) rounded to f32 */

typedef float v2f __attribute__((ext_vector_type(2)));
typedef float v8f __attribute__((ext_vector_type(8)));
typedef int   v4i __attribute__((ext_vector_type(4)));

#if defined(__HIP_DEVICE_COMPILE__) && \
    __has_builtin(__builtin_amdgcn_global_load_async_to_lds_b128)
#define USE_ASYNC_LDS 1
typedef __attribute__((address_space(1))) v4i* as1_v4i;
typedef __attribute__((address_space(3))) v4i* as3_v4i;
#endif

__device__ __forceinline__ float vertf(int i) { return -0.5f + (float)i * (1.0f / 63.0f); }
// pixel coord of normalized x/y for the 2D samples:
// ((v - mn)/(mx - mn + 1e-5)*2 - 1 + 1) * 127.5  with mn=-0.5, mx=0.5
__device__ __forceinline__ float xpix(int i) {
    return (vertf(i) + 0.5f) * (255.0f / 1.00001f);
}

// ---------------------------------------------------------------------------
// Kernel 0: build A[y][j] (256 x 64, y-major) into d_ws.
// A[j,y] = (1/64) * sum_k bilinear-y weights of sample row for (j,k).
// ---------------------------------------------------------------------------
__global__ __launch_bounds__(64)
void build_rot_weights(float* __restrict__ Aw) {
    __shared__ float acc[HW * V];           // 64 KB
    const int j = threadIdx.x;              // 0..63, owns column j
    for (int e = j; e < HW * V; e += V) acc[e] = 0.0f;
    __syncthreads();

    const float vy  = vertf(j);
    const float scl = 255.0f / (2.0f * SQ2H + 1e-5f);
    for (int k = 0; k < V; ++k) {
        float yr  = SQ2H * vy - SQ2H * vertf(k);      // rotated y-coordinate
        float Y   = (yr + SQ2H) * scl;                // pixel row in [0,255)
        float y0f = floorf(Y);
        int   y0  = (int)y0f;
        float wy  = Y - y0f;
        if (y0 >= 0 && y0 < HW)         acc[y0 * V + j]       += (1.0f - wy) * (1.0f / 64.0f);
        if (y0 + 1 >= 0 && y0 + 1 < HW) acc[(y0 + 1) * V + j] +=  wy         * (1.0f / 64.0f);
    }
    __syncthreads();
    for (int e = j; e < HW * V; e += V) Aw[e] = acc[e];
}

// ---------------------------------------------------------------------------
// Kernel 1: fused per-(b,c) pipeline.
//   Vbar = Vcc(gather of x_cc) + rowmix(x_mlo) @ A     (WMMA fp32 GEMM)
//   out  = Vbar                      (target==1)
//        = rot-bilinear-resample(Vbar, zero pad)       (target==0)
// ---------------------------------------------------------------------------
__global__ __launch_bounds__(256)
void fuse_project(const float* __restrict__ x_cc,
                  const float* __restrict__ x_mlo,
                  const int*   __restrict__ tgt,
                  const float* __restrict__ Aw,
                  float*       __restrict__ out) {
    __shared__ float sRM[HW * V];   // rowmix[y][i]   64 KB
    __shared__ float sAw[HW * V];   // A[y][j]        64 KB (staged once per WG)
    __shared__ float sV [V * V];    // Vbar[i][j]     16 KB

    const int bc  = blockIdx.x;     // 0..127  (b*16 + c)
    const int b   = bc >> 4;
    const int tid = threadIdx.x;
    const float* icc  = x_cc  + (size_t)bc * (HW * HW);
    const float* imlo = x_mlo + (size_t)bc * (HW * HW);

    // ---- Stage A-async: kick off Aw -> sAw copy (overlaps with A0/A1 math) ----
#ifdef USE_ASYNC_LDS
    #pragma unroll
    for (int n = 0; n < 16; ++n) {
        const int idx = (n * 256 + tid) * 4;            // float index, 16B aligned
        __builtin_amdgcn_global_load_async_to_lds_b128(
            (as1_v4i)(uintptr_t)(Aw + idx),
            (as3_v4i)(uint32_t)(uintptr_t)(&sAw[idx]),
            0, 0);
    }
#else
    for (int e = tid * 4; e < HW * V; e += 1024) {
        float4 v = *(const float4*)(Aw + e);
        *(float4*)(&sAw[e]) = v;
    }
#endif

    // ---- Stage A0: Vcc -> sV (4-tap bilinear gather, all taps in-bounds) ----
    for (int e = tid; e < V * V; e += 256) {
        int i = e >> 6, j = e & 63;
        float X = xpix(i), Y = xpix(j);
        float xf = floorf(X), yf = floorf(Y);
        int   x0 = (int)xf,   y0 = (int)yf;
        float wx = X - xf,    wy = Y - yf;
        int   x1 = min(x0 + 1, HW - 1), y1 = min(y0 + 1, HW - 1);
        float v00 = icc[y0 * HW + x0], v01 = icc[y0 * HW + x1];
        float v10 = icc[y1 * HW + x0], v11 = icc[y1 * HW + x1];
        sV[e] = (1.0f - wx) * (1.0f - wy) * v00 + wx * (1.0f - wy) * v01
              + (1.0f - wx) * wy          * v10 + wx * wy          * v11;
    }
    // ---- Stage A1: rowmix[y][i] = x-blend of x_mlo row y ----
    for (int n = 0; n < 64; ++n) {
        int e = n * 256 + tid;
        int y = e >> 6, i = e & 63;
        float X  = xpix(i);
        float xf = floorf(X);
        int   x0 = (int)xf;
        float wx = X - xf;
        int   x1 = min(x0 + 1, HW - 1);
        sRM[y * V + i] = (1.0f - wx) * imlo[y * HW + x0] + wx * imlo[y * HW + x1];
    }
#ifdef USE_ASYNC_LDS
#if __has_builtin(__builtin_amdgcn_s_wait_asynccnt)
    __builtin_amdgcn_s_wait_asynccnt(0);   // my async writes landed in LDS
#else
    asm volatile("s_wait_asynccnt 0x0" ::: "memory");
#endif
#endif
    __syncthreads();                        // everyone's stage-A data visible

    // ---- Stage B: Mbar[i][j] += sum_y sRM[y][i]*sAw[y][j], C-init from sV ----
    // 16 tiles of 16x16, 8 waves, 2 tiles each. fp32 WMMA, K-step 4.
    const int lane = tid & 31;
    const int half = lane >> 4;     // 0: lanes 0-15, 1: lanes 16-31
    const int l16  = lane & 15;
    const int wave = tid >> 5;
    for (int t = wave; t < 16; t += 8) {
        const int Mb = (t >> 2) << 4;
        const int Nb = (t & 3) << 4;
        v8f acc;
        #pragma unroll
        for (int r = 0; r < 8; ++r)
            acc[r] = sV[(Mb + r + half * 8) * V + (Nb + l16)];   // C/D layout

        const int arow = Mb + l16;   // A: lane -> M
        const int bcol = Nb + l16;   // B: lane -> N
        for (int y = 0; y < HW; y += 4) {
#if __has_builtin(__builtin_amdgcn_wmma_f32_16x16x4_f32)
            const int k = y + half * 2;                 // VGPR/half -> K
            v2f a, bf;
            a.x  = sRM[k * V + arow];  a.y  = sRM[(k + 1) * V + arow];
            bf.x = sAw[k * V + bcol];  bf.y = sAw[(k + 1) * V + bcol];
            acc = __builtin_amdgcn_wmma_f32_16x16x4_f32(
                      false, a, false, bf, (short)0, acc, false, false);
#else
            // exact scalar emulation of the same 16x16x4 step
            #pragma unroll
            for (int r = 0; r < 8; ++r) {
                const int M = Mb + r + half * 8;
                const int N = Nb + l16;
                float s = acc[r];
                #pragma unroll
                for (int kk = 0; kk < 4; ++kk)
                    s += sRM[(y + kk) * V + M] * sAw[(y + kk) * V + N];
                acc[r] = s;
            }
#endif
        }
        #pragma unroll
        for (int r = 0; r < 8; ++r)
            sV[(Mb + r + half * 8) * V + (Nb + l16)] = acc[r];
    }
    __syncthreads();

    // ---- Stage C: select projection and write out ----
    const int target = tgt[b];                         // target_indices[b,0]
    float* o = out + (size_t)bc * (V * V);
    const float th2 = 45.0f * 3.1415926f / 180.0f;     // reference's rot-grid theta
    const float c2  = cosf(th2), s2 = sinf(th2);
    for (int e = tid; e < V * V; e += 256) {
        float r;
        if (target == 1) {
            r = sV[e];                                 // proj_cc == Vbar
        } else {
            // proj_mlo: rotate (a=p, bi=q) then bilinear sample Vbar with zero pad
            int p = e >> 6, q = e & 63;
            float lp = -1.0f + (float)p * (2.0f / 63.0f);   // z source
            float lq = -1.0f + (float)q * (2.0f / 63.0f);   // y source
            float yp = c2 * lq - s2 * lp;              // -> H axis (j)
            float zp = s2 * lq + c2 * lp;              // -> D axis (i)
            float Zf = (zp + 1.0f) * 31.5f;
            float Yf = (yp + 1.0f) * 31.5f;
            float z0f = floorf(Zf), y0f = floorf(Yf);
            int   z0  = (int)z0f,   y0  = (int)y0f;
            float wz = Zf - z0f,    wy = Yf - y0f;
            r = 0.0f;
            #pragma unroll
            for (int dz = 0; dz < 2; ++dz) {
                int zi = z0 + dz;
                if (zi < 0 || zi > V - 1) continue;
                float wzz = dz ? wz : (1.0f - wz);
                #pragma unroll
                for (int dy = 0; dy < 2; ++dy) {
                    int yi = y0 + dy;
                    if (yi < 0 || yi > V - 1) continue;
                    float w = wzz * (dy ? wy : (1.0f - wy));
                    r += w * sV[zi * V + yi];
                }
            }
        }
        o[e] = r;
    }
}

// ---------------------------------------------------------------------------
extern "C" void kernel_launch(void* const* d_in, const int* in_sizes, int n_in,
                              void* d_out, int out_size, void* d_ws, size_t ws_size,
                              hipStream_t stream) {
    (void)in_sizes; (void)n_in; (void)out_size; (void)ws_size;
    const float* x_cc  = (const float*)d_in[0];
    const float* x_mlo = (const float*)d_in[1];
    const int*   tgt   = (const int*)d_in[2];
    float*       Aw    = (float*)d_ws;          // 256*64 floats = 64 KB scratch
    float*       outp  = (float*)d_out;

    build_rot_weights<<<1, 64, 0, stream>>>(Aw);
    fuse_project<<<128, 256, 0, stream>>>(x_cc, x_mlo, tgt, Aw, outp);
}